// MMGNet_27066883899996
// MI455X (gfx1250) — compile-verified
//
#include <hip/hip_runtime.h>
#include <hip/hip_bf16.h>
#include <math.h>

typedef __attribute__((ext_vector_type(2))) float v2f;
typedef __attribute__((ext_vector_type(8))) float v8f;

#define H_DIM 256
#define NFILT 8
#define TILE_P 64
#define KCH 32

// ---------------------------------------------------------------------------
// Kernel 1: KNN (k=8) over cond_bank + IDW-weighted latent gather -> ws
// ---------------------------------------------------------------------------
__global__ void mmg_knn_idw(const float* __restrict__ cond,
                            const float* __restrict__ bank,
                            const float* __restrict__ latents,
                            float* __restrict__ latOut, int M, int L) {
  const int b = blockIdx.x, tid = threadIdx.x;   // 256 threads
  __shared__ float cb[16];
  __shared__ float sD[2048];
  __shared__ int   sI[2048];
  __shared__ float selW[8];
  __shared__ int   selI[8];
  if (tid < 16) cb[tid] = cond[b * 16 + tid];
  __syncthreads();
  float c[16];
#pragma unroll
  for (int j = 0; j < 16; ++j) c[j] = cb[j];
  float bd[8]; int bi[8];
#pragma unroll
  for (int k = 0; k < 8; ++k) { bd[k] = 3.4e38f; bi[k] = 0; }
  for (int m = tid; m < M; m += 256) {
    float d2 = 0.f;
#pragma unroll
    for (int j = 0; j < 16; ++j) {
      float df = c[j] - bank[m * 16 + j];
      d2 = fmaf(df, df, d2);
    }
    if (d2 < bd[7]) {                 // per-thread insertion top-8 (by d^2)
      int k = 7;
      while (k > 0 && bd[k - 1] > d2) { bd[k] = bd[k - 1]; bi[k] = bi[k - 1]; --k; }
      bd[k] = d2; bi[k] = m;
    }
  }
#pragma unroll
  for (int k = 0; k < 8; ++k) { sD[tid * 8 + k] = bd[k]; sI[tid * 8 + k] = bi[k]; }
  __syncthreads();
  if (tid == 0) {                     // final top-8 of 2048 candidates + IDW
    float wsum = 0.f;
    for (int k = 0; k < 8; ++k) {
      float best = 3.4e38f; int bj = 0;
      for (int j = 0; j < 2048; ++j)
        if (sD[j] < best) { best = sD[j]; bj = j; }
      sD[bj] = 3.4e38f;
      float d = sqrtf(fmaxf(best, 0.f));
      float w = 1.f / (d + 1e-8f);
      selW[k] = w; selI[k] = sI[bj]; wsum += w;
    }
    float inv = 1.f / wsum;
    for (int k = 0; k < 8; ++k) selW[k] *= inv;
  }
  __syncthreads();
  if (tid < L) {                      // L == 256
    float a = 0.f;
#pragma unroll
    for (int k = 0; k < 8; ++k)
      a = fmaf(selW[k], latents[(long)selI[k] * L + tid], a);
    latOut[b * L + tid] = a;
  }
}

// ---------------------------------------------------------------------------
// Kernel 2: per-(layer,batch) latent-fuse constants: latC = latent@B_i^T + bias_i
// (fuse_A0 * (x*0) == 0 in the reference, so layer 0 reduces to this too)
// ---------------------------------------------------------------------------
__global__ void mmg_latc(const float* __restrict__ lat,
                         const float* __restrict__ fB,
                         const float* __restrict__ fbias,
                         float* __restrict__ latC, int L) {
  const int b = blockIdx.x, i = blockIdx.y, o = threadIdx.x;  // 256 threads
  __shared__ float lv[H_DIM];
  lv[o] = lat[b * L + o];
  __syncthreads();
  const float* Brow = fB + ((long)i * H_DIM + o) * L;
  float s = fbias[i * H_DIM + o];
#pragma unroll 4
  for (int j = 0; j < H_DIM; ++j) s = fmaf(lv[j], Brow[j], s);
  latC[((long)i * gridDim.x + b) * H_DIM + o] = s;
}

// ---------------------------------------------------------------------------
// Kernel 3: fused 8-layer Gabor network. zi tile kept in WMMA C/D-layout
// registers; GEMM via V_WMMA_F32_16X16X4_F32, A_i streamed through LDS.
// ---------------------------------------------------------------------------
__device__ __forceinline__ float gabor_val(float4 xv, float w0, float w1, float w2,
                                           float m0, float m1, float m2,
                                           float mu2, float bbv, float ggv) {
  float arg = fmaf(xv.x, w0, fmaf(xv.y, w1, fmaf(xv.z, w2, bbv)));
  float xm  = fmaf(xv.x, m0, fmaf(xv.y, m1, xv.z * m2));
  float dd  = (xv.w + mu2 - 2.f * xm) * ggv;
  return __sinf(arg) * __expf(-0.5f * dd);
}

__global__ __launch_bounds__(128)
void mmg_main(const float* __restrict__ x,
              const float* __restrict__ gW, const float* __restrict__ gB,
              const float* __restrict__ gMu, const float* __restrict__ gGa,
              const float* __restrict__ fA, const float* __restrict__ latC,
              const float* __restrict__ outW, const float* __restrict__ outB,
              float* __restrict__ out, int P) {
  __shared__ float  ziLDS[TILE_P * H_DIM];   // 64 KB: zi tile (A-operand source)
  __shared__ float  aLDS[H_DIM * KCH];       // 32 KB: fuse_A K-chunk (B-operand)
  __shared__ float4 xLDS[TILE_P];
  __shared__ float  outLDS[TILE_P];

  const int tid  = threadIdx.x;
  const int lane = tid & 31;
  const int wave = tid >> 5;
  const int half = lane >> 4;
  const int l15  = lane & 15;
  const int bb_  = blockIdx.y;
  const int tileBase = blockIdx.x * TILE_P;

  if (tid < TILE_P) {
    const float* xp = x + ((long)bb_ * P + tileBase + tid) * 3;
    float x0 = xp[0], x1 = xp[1], x2 = xp[2];
    xLDS[tid] = make_float4(x0, x1, x2, x0 * x0 + x1 * x1 + x2 * x2);
  }
  __syncthreads();

  // Each lane covers rows {rowBase+r, r=0..7} of the 64-row tile (C/D layout:
  // VGPR r, lanes 0-15 -> M=r, lanes 16-31 -> M=r+8; 4 waves x 16-row stripes).
  const int rowBase = wave * 16 + half * 8;

  v8f acc[16];                                 // 16 col-tiles x (16x16 f32)

  { // ---- layer 0: zi = gabor0(x) * latC0 (coord term is exactly zero) ----
    float4 xv[8];                              // short-lived: re-read from LDS
#pragma unroll
    for (int r = 0; r < 8; ++r) xv[r] = xLDS[rowBase + r];
    const float* lcp = latC + (long)bb_ * H_DIM;   // layer 0 slice
#pragma unroll
    for (int t = 0; t < 16; ++t) {
      const int o = t * 16 + l15;
      const float w0 = gW[o*3+0], w1 = gW[o*3+1], w2 = gW[o*3+2];
      const float m0 = gMu[o*3+0], m1 = gMu[o*3+1], m2 = gMu[o*3+2];
      const float mu2 = m0*m0 + m1*m1 + m2*m2;
      const float bbv = gB[o], ggv = gGa[o];
      const float lc  = lcp[o];
#pragma unroll
      for (int r = 0; r < 8; ++r)
        acc[t][r] = lc * gabor_val(xv[r], w0, w1, w2, m0, m1, m2, mu2, bbv, ggv);
    }
  }

#pragma unroll 1
  for (int layer = 1; layer < NFILT; ++layer) {
    // 1) park zi in LDS (source of WMMA A-operands this layer)
#pragma unroll
    for (int t = 0; t < 16; ++t)
#pragma unroll
      for (int r = 0; r < 8; ++r)
        ziLDS[(rowBase + r) * H_DIM + t * 16 + l15] = acc[t][r];
    __syncthreads();

    // 2) re-seed accumulators with per-batch latent constant for this layer
    const float* lcp = latC + ((long)layer * gridDim.y + bb_) * H_DIM;
#pragma unroll
    for (int t = 0; t < 16; ++t) {
      const float lc = lcp[t * 16 + l15];
#pragma unroll
      for (int r = 0; r < 8; ++r) acc[t][r] = lc;
    }

    // 3) acc += zi @ fuse_A[layer-1]^T, K streamed through LDS in 32-chunks
    const float* Al  = fA + (long)(layer - 1) * H_DIM * H_DIM;
    const int rowA = wave * 16 + l15;          // A-operand row (M = lane&15)
#pragma unroll 1
    for (int ko = 0; ko < H_DIM / KCH; ++ko) {
#pragma unroll 4
      for (int j = 0; j < 16; ++j) {           // 128 thr * 16 float4 = 256x32 f32
        const int linear = j * 128 + tid;
        const int o  = linear >> 3;
        const int c4 = (linear & 7) * 4;
        *(float4*)(aLDS + o * KCH + c4) =
            *(const float4*)(Al + o * H_DIM + ko * KCH + c4);
      }
      if (ko + 1 < H_DIM / KCH)                // pull next chunk toward the WGP
        __builtin_prefetch(Al + (tid * 2) * H_DIM + (ko + 1) * KCH, 0, 1);
      __syncthreads();

      // 16x4 f32 operand layout: VGPR0/1 hold K,K+1 with K = k0 + 2*half,
      // so each operand is one contiguous ds_load_b64 per lane.
      const float* abase = ziLDS + rowA * H_DIM + ko * KCH + 2 * half;
#pragma unroll
      for (int t = 0; t < 16; ++t) {
        const float* bbase = aLDS + (t * 16 + l15) * KCH + 2 * half;
#pragma unroll
        for (int kc = 0; kc < KCH / 4; ++kc) {
          v2f a2 = *(const v2f*)(abase + kc * 4);
          v2f b2 = *(const v2f*)(bbase + kc * 4);
          acc[t] = __builtin_amdgcn_wmma_f32_16x16x4_f32(
              false, a2, false, b2, (short)0, acc[t], false, false);
        }
      }
      __syncthreads();
    }

    // 4) gate with Gabor filter `layer`
    {
      float4 xv[8];                            // short-lived reload from LDS
#pragma unroll
      for (int r = 0; r < 8; ++r) xv[r] = xLDS[rowBase + r];
      const float* gWl = gW  + layer * H_DIM * 3;
      const float* gMl = gMu + layer * H_DIM * 3;
      const float* gBl = gB  + layer * H_DIM;
      const float* gGl = gGa + layer * H_DIM;
#pragma unroll
      for (int t = 0; t < 16; ++t) {
        const int o = t * 16 + l15;
        const float w0 = gWl[o*3+0], w1 = gWl[o*3+1], w2 = gWl[o*3+2];
        const float m0 = gMl[o*3+0], m1 = gMl[o*3+1], m2 = gMl[o*3+2];
        const float mu2 = m0*m0 + m1*m1 + m2*m2;
        const float bbv = gBl[o], ggv = gGl[o];
#pragma unroll
        for (int r = 0; r < 8; ++r)
          acc[t][r] *= gabor_val(xv[r], w0, w1, w2, m0, m1, m2, mu2, bbv, ggv);
      }
    }
  }

  // out = zi @ out_W^T + out_b. Deterministic wave32 shuffle reduce per row.
  float rp[8];
#pragma unroll
  for (int r = 0; r < 8; ++r) rp[r] = 0.f;
#pragma unroll
  for (int t = 0; t < 16; ++t) {
    const float w = outW[t * 16 + l15];
#pragma unroll
    for (int r = 0; r < 8; ++r) rp[r] = fmaf(acc[t][r], w, rp[r]);
  }
#pragma unroll
  for (int r = 0; r < 8; ++r) {
    float v = rp[r];
#pragma unroll
    for (int s = 8; s >= 1; s >>= 1) v += __shfl_xor(v, s, 16);
    if (l15 == 0) outLDS[rowBase + r] = v;     // unique row per (wave,half,r)
  }
  __syncthreads();
  if (tid < TILE_P)
    out[(long)bb_ * P + tileBase + tid] = outLDS[tid] + outB[0];
}

// ---------------------------------------------------------------------------
extern "C" void kernel_launch(void* const* d_in, const int* in_sizes, int n_in,
                              void* d_out, int out_size, void* d_ws, size_t ws_size,
                              hipStream_t stream) {
  const float* x       = (const float*)d_in[0];
  const float* cond    = (const float*)d_in[1];
  const float* bank    = (const float*)d_in[2];
  const float* latents = (const float*)d_in[3];
  const float* gW      = (const float*)d_in[4];
  const float* gB      = (const float*)d_in[5];
  const float* gMu     = (const float*)d_in[6];
  const float* gGa     = (const float*)d_in[7];
  // d_in[8] (fuse_A0) multiplies zeroed coords in the reference: dead input.
  const float* fA      = (const float*)d_in[9];
  const float* fB      = (const float*)d_in[10];
  const float* fbias   = (const float*)d_in[11];
  const float* outW    = (const float*)d_in[12];
  const float* outB    = (const float*)d_in[13];
  // d_in[14] = cond_knn (8), matched by the compile-time top-8 path.

  const int Bc = in_sizes[1] / 16;            // 8
  const int M  = in_sizes[2] / 16;            // 100000
  const int L  = in_sizes[3] / M;             // 256
  const int P  = in_sizes[0] / (Bc * 3);      // 16384

  float* wsLat  = (float*)d_ws;               // [Bc][L]
  float* wsLatC = wsLat + Bc * L;             // [NFILT][Bc][H]

  mmg_knn_idw<<<Bc, 256, 0, stream>>>(cond, bank, latents, wsLat, M, L);
  mmg_latc<<<dim3(Bc, NFILT), 256, 0, stream>>>(wsLat, fB, fbias, wsLatC, L);
  mmg_main<<<dim3(P / TILE_P, Bc), 128, 0, stream>>>(
      x, gW, gB, gMu, gGa, fA, wsLatC, outW, outB, (float*)d_out, P);
}